// ProxyNcaLayer_5729486373477
// MI455X (gfx1250) — compile-verified
//
#include <hip/hip_runtime.h>

// ProxyNCA loss + EMA center update for MI455X (gfx1250, wave32).
// M=4096 samples, C=512 classes, D=128 dims.
//
// Pipeline:
//  1) prep:   per-row label extraction, f32 atomic counts, residual R = centers[label] - x0
//  2) delta:  delta = onehot^T @ R as a real f32 WMMA GEMM (V_WMMA_F32_16X16X4_F32),
//             dual accumulators to pipeline the matrix unit, EMA update fused
//  3) loss:   L1 distance matrix on VALU (not expressible as matmul) with
//             LDS-tiled x rows (float4 DS broadcasts), b128 center loads,
//             stable logsumexp (max term is provably 0 since d>=0)

#define M_N   4096
#define C_N   512
#define D_N   128
#define ALPHAF 0.5f
#define ROWS  32   // rows of x0 per block in the loss kernel

typedef __attribute__((ext_vector_type(2))) float v2f;
typedef __attribute__((ext_vector_type(8))) float v8f;

// ---------------------------------------------------------------------------
// Kernel 1: labels, counts, residual R[i,:] = centers[label_i,:] - x0[i,:]
// ---------------------------------------------------------------------------
__global__ __launch_bounds__(128)
void pnca_prep_kernel(const float* __restrict__ x0,
                      const float* __restrict__ onehot,
                      const float* __restrict__ centers,
                      int* __restrict__ labels,
                      float* __restrict__ counts,
                      float* __restrict__ R) {
    __shared__ int slab;
    const int i = blockIdx.x;
    const int t = threadIdx.x;

    // exactly one entry per row is 1.0
    for (int c = t; c < C_N; c += 128) {
        if (onehot[i * C_N + c] > 0.5f) slab = c;
    }
    __syncthreads();
    const int lab = slab;

    if (t == 0) {
        labels[i] = lab;
        atomicAdd(&counts[lab], 1.0f);  // exact: small integer values in f32
    }
    // one thread per D element (128 threads == D_N)
    R[i * D_N + t] = centers[lab * D_N + t] - x0[i * D_N + t];
}

// ---------------------------------------------------------------------------
// Kernel 2: delta = onehot^T (C x M) @ R (M x D) via V_WMMA_F32_16X16X4_F32.
// One wave per 16x16 output tile; K loop over 4096 samples, two independent
// accumulator chains so consecutive v_wmma ops pipeline instead of stalling
// on the SRC2 RAW hazard (ISA 7.12.1).
// Fused: out_centers = centers - ALPHA * delta / (counts + 1)
//
// A-matrix (16x4, MxK) VGPR layout (ISA 7.12.2): lanes 0-15 hold K=0,1 in
// v0,v1; lanes 16-31 hold K=2,3. B (4x16, KxN) mirrors it. C/D (16x16):
// VGPR r -> M=r (lanes 0-15) / M=r+8 (lanes 16-31), N = lane & 15.
// ---------------------------------------------------------------------------
__global__ __launch_bounds__(32)
void pnca_delta_wmma_kernel(const float* __restrict__ onehot,
                            const float* __restrict__ R,
                            const float* __restrict__ centers,
                            const float* __restrict__ counts,
                            float* __restrict__ out_centers) {
    const int tile = blockIdx.x;        // 0..255 = (C/16)*(D/16)
    const int c0   = (tile >> 3) << 4;  // 32 class tiles
    const int d0   = (tile & 7) << 4;   // 8  dim tiles
    const int lane = threadIdx.x;
    const int half = lane >> 4;         // 0: K pair {0,1}, 1: K pair {2,3}
    const int l    = lane & 15;

    v8f acc0 = {0.f, 0.f, 0.f, 0.f, 0.f, 0.f, 0.f, 0.f};
    v8f acc1 = {0.f, 0.f, 0.f, 0.f, 0.f, 0.f, 0.f, 0.f};

    const float* __restrict__ Abase = onehot + c0 + l;  // A[m][k] = onehot[k*C + c0+m]
    const float* __restrict__ Bbase = R + d0 + l;       // B[k][n] = R[k*D + d0+n]

#pragma unroll 4
    for (int k0 = 0; k0 < M_N; k0 += 8) {
        const int kA = k0 + half * 2;
        v2f a0, b0, a1, b1;
        a0.x = Abase[kA * C_N];
        a0.y = Abase[(kA + 1) * C_N];
        b0.x = Bbase[kA * D_N];
        b0.y = Bbase[(kA + 1) * D_N];
        a1.x = Abase[(kA + 4) * C_N];
        a1.y = Abase[(kA + 5) * C_N];
        b1.x = Bbase[(kA + 4) * D_N];
        b1.y = Bbase[(kA + 5) * D_N];
        // 8 args: (neg_a, A, neg_b, B, c_mod, C, reuse_a, reuse_b)
        acc0 = __builtin_amdgcn_wmma_f32_16x16x4_f32(
            false, a0, false, b0, (short)0, acc0, false, false);
        acc1 = __builtin_amdgcn_wmma_f32_16x16x4_f32(
            false, a1, false, b1, (short)0, acc1, false, false);
    }

    // Fused EMA center update on the accumulator tile.
#pragma unroll
    for (int r = 0; r < 8; ++r) {
        const int cc = c0 + r + half * 8;
        const int dd = d0 + l;
        const float cnt = counts[cc];
        const float cen = centers[cc * D_N + dd];
        const float delta = acc0[r] + acc1[r];
        out_centers[cc * D_N + dd] = cen - ALPHAF * delta / (cnt + 1.0f);
    }
}

// ---------------------------------------------------------------------------
// Kernel 3: L1 distances + loss. 512 threads = one per class (single pass).
// Each block owns 32 rows of x0 in LDS (16 KB); x read as float4 broadcasts
// (ds_load_b128), centers as float4 rows (global_load_b128). d >= 0 means
// the logsumexp max is exactly the zeroed positive slot: no max reduction.
// ---------------------------------------------------------------------------
__global__ __launch_bounds__(512)
void pnca_loss_kernel(const float* __restrict__ x0,
                      const float* __restrict__ centers,
                      const int* __restrict__ labels,
                      float* __restrict__ loss) {
    __shared__ float sx[ROWS * D_N];   // 16 KB of x rows, float4-aligned
    __shared__ float wsum[16];         // per-wave partials (16 waves of 32)
    __shared__ float sdpos;

    const int t  = threadIdx.x;        // class index 0..511
    const int r0 = blockIdx.x * ROWS;

    // coalesced float4 fill of the x tile
    {
        const float4* __restrict__ src = (const float4*)(x0 + r0 * D_N);
        float4* dst = (float4*)sx;
        for (int idx = t; idx < ROWS * D_N / 4; idx += 512)
            dst[idx] = src[idx];
    }
    __syncthreads();

    float acc[ROWS];
#pragma unroll
    for (int r = 0; r < ROWS; ++r) acc[r] = 0.0f;

    const float* __restrict__ crow = centers + t * D_N;
    __builtin_prefetch(crow, 0, 0);    // global_prefetch_b8

    for (int k4 = 0; k4 < D_N / 4; ++k4) {
        const float4 cv = ((const float4*)crow)[k4];   // global_load_b128
#pragma unroll
        for (int r = 0; r < ROWS; ++r) {
            const float4 xv = ((const float4*)(sx + r * D_N))[k4];  // ds_load_b128 broadcast
            acc[r] += fabsf(xv.x - cv.x) + fabsf(xv.y - cv.y) +
                      fabsf(xv.z - cv.z) + fabsf(xv.w - cv.w);
        }
    }

    const int wave = t >> 5;
    for (int r = 0; r < ROWS; ++r) {
        const int lab = labels[r0 + r];             // uniform load
        float e = (t == lab) ? 1.0f : __expf(-acc[r]);
        if (t == lab) sdpos = acc[r];
        // wave32 butterfly reduction
        for (int off = 16; off >= 1; off >>= 1)
            e += __shfl_xor(e, off, 32);
        if ((t & 31) == 0) wsum[wave] = e;
        __syncthreads();
        if (t == 0) {
            float s = 0.0f;
#pragma unroll
            for (int w = 0; w < 16; ++w) s += wsum[w];
            loss[r0 + r] = sdpos + __logf(s);       // -s_pos + logsumexp (max=0)
        }
        __syncthreads();
    }
}

// ---------------------------------------------------------------------------
// Host launch
// ---------------------------------------------------------------------------
extern "C" void kernel_launch(void* const* d_in, const int* in_sizes, int n_in,
                              void* d_out, int out_size, void* d_ws, size_t ws_size,
                              hipStream_t stream) {
    const float* x0      = (const float*)d_in[0];  // (M, D)
    const float* onehot  = (const float*)d_in[1];  // (M, C)
    const float* centers = (const float*)d_in[2];  // (C, D)

    float* out         = (float*)d_out;
    float* loss        = out;          // M floats
    float* out_centers = out + M_N;    // C*D floats

    char*  ws     = (char*)d_ws;
    float* counts = (float*)ws;                         // 512 f32   (2 KB)
    int*   labels = (int*)(ws + 2048);                  // 4096 i32  (16 KB)
    float* R      = (float*)(ws + 2048 + 16384);        // M*D f32   (2 MB)

    hipMemsetAsync(counts, 0, C_N * sizeof(float), stream);

    pnca_prep_kernel<<<M_N, 128, 0, stream>>>(x0, onehot, centers, labels, counts, R);
    pnca_delta_wmma_kernel<<<(C_N / 16) * (D_N / 16), 32, 0, stream>>>(
        onehot, R, centers, counts, out_centers);
    pnca_loss_kernel<<<M_N / ROWS, 512, 0, stream>>>(x0, centers, labels, loss);
}